// GAT_30434138260191
// MI455X (gfx1250) — compile-verified
//
#include <hip/hip_runtime.h>
#include <math.h>

// GAT on MI455X (gfx1250): fp32 WMMA (16x16x4) for the dense projections,
// async global->LDS DMA (ASYNCcnt path) for weight-panel staging, and a
// memory-lean 3-pass edge softmax/scatter for the message passing (the true
// bottleneck at ~0.9 GB of gather+scatter traffic per conv vs 23.3 TB/s HBM).

#define N_NODESK 50000
#define N_EDGESK 800000
#define HEADSK   4
#define HIDK     32
#define HCK      128
#define ETOTK    (N_EDGESK + N_NODESK)

typedef float v2f __attribute__((ext_vector_type(2)));
typedef float v8f __attribute__((ext_vector_type(8)));

__device__ __forceinline__ unsigned ford_enc(float f) {
  unsigned u = __float_as_uint(f);
  return (u & 0x80000000u) ? ~u : (u | 0x80000000u);
}
__device__ __forceinline__ float ford_dec(unsigned u) {
  return (u & 0x80000000u) ? __uint_as_float(u & 0x7fffffffu) : __uint_as_float(~u);
}

// ---------------- WMMA GEMM: C[M,NCOL] = A[M,128] * B[128,NCOL] (+bias) ----
// Block = 256 threads = 8 wave32s. Full B panel staged in LDS via async
// global->LDS DMA (no VGPR round trip). Each wave owns one 16-row tile and
// all NCOL/16 column tiles (8 v8f accumulators for HC=128).
template <int NCOL, bool BIAS>
__global__ void gemm_wmma_kernel(const float* __restrict__ A,
                                 const float* __restrict__ B,
                                 const float* __restrict__ bias,
                                 float* __restrict__ C, int M) {
  __shared__ float Bs[128 * NCOL];

  // Async DMA stage: each of the 256 threads copies 16B per issue.
  {
    const unsigned lds0 = (unsigned)(uintptr_t)(&Bs[0]);  // low 32 bits = LDS byte addr
    const unsigned long long gB = (unsigned long long)(uintptr_t)B;
    const unsigned off0 = threadIdx.x * 16u;
    constexpr unsigned BYTES = 128u * NCOL * 4u;
#pragma unroll
    for (unsigned it = 0; it < BYTES / (256u * 16u); ++it) {
      unsigned go = off0 + it * 256u * 16u;   // byte offset into B
      unsigned lo = lds0 + go;                // LDS byte address
      asm volatile("global_load_async_to_lds_b128 %0, %1, %2"
                   :: "v"(lo), "v"(go), "s"(gB)
                   : "memory");
    }
    asm volatile("s_wait_asynccnt 0x0" ::: "memory");
  }
  __syncthreads();

  const int wave = threadIdx.x >> 5;   // wave32
  const int lane = threadIdx.x & 31;
  const int tile = blockIdx.x * 8 + wave;
  const int rowTiles = M >> 4;         // M multiple of 16
  if (tile >= rowTiles) return;

  const int mrow  = tile * 16 + (lane & 15);
  const int khalf = (lane >> 4) << 1;  // lanes 0-15 -> K+0/K+1 ; 16-31 -> K+2/K+3
  const int ncol0 = lane & 15;

  constexpr int CT = NCOL / 16;
  v8f acc[CT] = {};

  const float* arow = A + (size_t)mrow * 128;
#pragma unroll 4
  for (int kk = 0; kk < 128; kk += 4) {
    v2f a;
    a.x = arow[kk + khalf];
    a.y = arow[kk + khalf + 1];
    const int kb = kk + khalf;
#pragma unroll
    for (int t = 0; t < CT; ++t) {
      v2f b;
      b.x = Bs[kb * NCOL + t * 16 + ncol0];
      b.y = Bs[(kb + 1) * NCOL + t * 16 + ncol0];
      acc[t] = __builtin_amdgcn_wmma_f32_16x16x4_f32(
          /*neg_a=*/false, a, /*neg_b=*/false, b,
          /*c_mod=*/(short)0, acc[t], /*reuse_a=*/false, /*reuse_b=*/false);
    }
  }

  const int rbase = tile * 16 + ((lane >> 4) << 3);  // lanes 16-31 hold M=8..15
#pragma unroll
  for (int t = 0; t < CT; ++t) {
    const int c = t * 16 + ncol0;
    const float bv = BIAS ? bias[c] : 0.f;
#pragma unroll
    for (int i = 0; i < 8; ++i) {
      C[(size_t)(rbase + i) * NCOL + c] = acc[t][i] + bv;
    }
  }
}

// ---------------- small helper kernels ------------------------------------
__global__ void edge_sum_kernel(const float* __restrict__ ew, float* __restrict__ out) {
  __shared__ float sdata[256];
  float s = 0.f;
  for (int i = blockIdx.x * blockDim.x + threadIdx.x; i < N_EDGESK;
       i += gridDim.x * blockDim.x)
    s += ew[i];
  sdata[threadIdx.x] = s;
  __syncthreads();
  for (int off = 128; off > 0; off >>= 1) {
    if ((int)threadIdx.x < off) sdata[threadIdx.x] += sdata[threadIdx.x + off];
    __syncthreads();
  }
  if (threadIdx.x == 0) atomicAdd(out, sdata[0]);
}

// s[h] = sum_c W_edge[h*32+c] * att_edge[h*32+c]   (a_edge = ew * s[h])
__global__ void prep_kernel(const float* __restrict__ W_edge,
                            const float* __restrict__ att_edge,
                            float* __restrict__ scal) {
  int h = threadIdx.x >> 5, c = threadIdx.x & 31;
  float p = W_edge[h * 32 + c] * att_edge[h * 32 + c];
  for (int off = 16; off > 0; off >>= 1) p += __shfl_down(p, off, 32);
  if (c == 0) scal[h] = p;
}

__global__ void node_att_kernel(const float* __restrict__ h,
                                const float* __restrict__ att_src,
                                const float* __restrict__ att_dst,
                                float* __restrict__ a_src,
                                float* __restrict__ a_dst) {
  int tid = blockIdx.x * blockDim.x + threadIdx.x;
  if (tid >= N_NODESK * HEADSK) return;
  int n = tid >> 2, hh = tid & 3;
  const float* hp = h + (size_t)n * HCK + hh * HIDK;
  const float* as = att_src + hh * HIDK;
  const float* ad = att_dst + hh * HIDK;
  float ssum = 0.f, dsum = 0.f;
#pragma unroll
  for (int c = 0; c < HIDK; ++c) {
    float v = hp[c];
    ssum += v * as[c];
    dsum += v * ad[c];
  }
  a_src[tid] = ssum;
  a_dst[tid] = dsum;
}

__global__ void edge_alpha_kernel(const int* __restrict__ srcs,
                                  const int* __restrict__ dsts,
                                  const float* __restrict__ ew,
                                  const float* __restrict__ scal,
                                  const float* __restrict__ meansum, float inv_e,
                                  int use_mean,
                                  const float* __restrict__ a_src,
                                  const float* __restrict__ a_dst,
                                  float* __restrict__ alpha,
                                  unsigned* __restrict__ amax) {
  int e = blockIdx.x * blockDim.x + threadIdx.x;
  if (e >= ETOTK) return;
  int s, d;
  float ea;
  if (e < N_EDGESK) {
    s = srcs[e];
    d = dsts[e];
    ea = ew[e];
  } else {
    s = d = e - N_EDGESK;
    ea = use_mean ? meansum[0] * inv_e : 0.f;
  }
  const float4 as4 = *(const float4*)(a_src + 4 * (size_t)s);
  const float4 ad4 = *(const float4*)(a_dst + 4 * (size_t)d);
  float asv[4] = {as4.x, as4.y, as4.z, as4.w};
  float adv[4] = {ad4.x, ad4.y, ad4.z, ad4.w};
#pragma unroll
  for (int hh = 0; hh < 4; ++hh) {
    float al = asv[hh] + adv[hh] + ea * scal[hh];
    al = al > 0.f ? al : 0.2f * al;  // LeakyReLU(0.2)
    alpha[(size_t)e * 4 + hh] = al;
    atomicMax(&amax[(size_t)d * 4 + hh], ford_enc(al));
  }
}

__global__ void edge_exp_kernel(const int* __restrict__ dsts,
                                float* __restrict__ alpha,
                                const unsigned* __restrict__ amax,
                                float* __restrict__ denom) {
  int e = blockIdx.x * blockDim.x + threadIdx.x;
  if (e >= ETOTK) return;
  int d = (e < N_EDGESK) ? dsts[e] : e - N_EDGESK;
#pragma unroll
  for (int hh = 0; hh < 4; ++hh) {
    float am = ford_dec(amax[(size_t)d * 4 + hh]);
    float ex = __expf(alpha[(size_t)e * 4 + hh] - am);
    alpha[(size_t)e * 4 + hh] = ex;
    atomicAdd(&denom[(size_t)d * 4 + hh], ex);
  }
}

// one thread per (edge, head): 32-wide weighted scatter-add of h[src] into acc[dst]
__global__ void edge_scatter_kernel(const int* __restrict__ srcs,
                                    const int* __restrict__ dsts,
                                    const float* __restrict__ eexp,
                                    const float* __restrict__ denom,
                                    const float* __restrict__ h,
                                    float* __restrict__ acc) {
  int tid = blockIdx.x * blockDim.x + threadIdx.x;
  if (tid >= ETOTK * HEADSK) return;
  int e = tid >> 2, hh = tid & 3;
  int s, d;
  if (e < N_EDGESK) {
    s = srcs[e];
    d = dsts[e];
  } else {
    s = d = e - N_EDGESK;
  }
  float w = eexp[(size_t)e * 4 + hh] / (denom[(size_t)d * 4 + hh] + 1e-16f);
  const float* hs = h + (size_t)s * HCK + hh * HIDK;
  float* ao = acc + (size_t)d * HCK + hh * HIDK;
#pragma unroll
  for (int c = 0; c < HIDK; c += 4) {
    float4 v = *(const float4*)(hs + c);
    atomicAdd(ao + c + 0, v.x * w);
    atomicAdd(ao + c + 1, v.y * w);
    atomicAdd(ao + c + 2, v.z * w);
    atomicAdd(ao + c + 3, v.w * w);
  }
}

// out = relu(bn(acc + bias))
__global__ void finalize_kernel(const float* __restrict__ acc,
                                const float* __restrict__ bias,
                                const float* __restrict__ gamma,
                                const float* __restrict__ beta,
                                const float* __restrict__ rm,
                                const float* __restrict__ rv,
                                float* __restrict__ xout) {
  int tid = blockIdx.x * blockDim.x + threadIdx.x;
  if (tid >= N_NODESK * HCK) return;
  int c = tid & (HCK - 1);
  float v = acc[tid] + bias[c];
  v = (v - rm[c]) * (gamma[c] * rsqrtf(rv[c] + 1e-5f)) + beta[c];
  xout[tid] = v > 0.f ? v : 0.f;
}

__global__ void combine_kernel(const float* __restrict__ a,
                               const float* __restrict__ b,
                               float* __restrict__ o) {
  int tid = blockIdx.x * blockDim.x + threadIdx.x;
  if (tid >= N_NODESK * HCK) return;
  float v = a[tid] + b[tid];
  o[tid] = v > 0.f ? v : 0.f;
}

// ---------------- host side ------------------------------------------------
extern "C" void kernel_launch(void* const* d_in, const int* in_sizes, int n_in,
                              void* d_out, int out_size, void* d_ws, size_t ws_size,
                              hipStream_t stream) {
  (void)in_sizes; (void)n_in; (void)out_size; (void)ws_size;
  const float* x   = (const float*)d_in[0];
  const int*   ei  = (const int*)d_in[1];
  const int*   srcs = ei;
  const int*   dsts = ei + N_EDGESK;
  const float* ew  = (const float*)d_in[2];

  struct Conv { const float *W, *att_src, *att_dst, *W_edge, *att_edge, *bias; };
  struct Bn   { const float *gamma, *beta, *rm, *rv; };
  auto conv_at = [&](int b) {
    return Conv{(const float*)d_in[b],     (const float*)d_in[b + 1],
                (const float*)d_in[b + 2], (const float*)d_in[b + 3],
                (const float*)d_in[b + 4], (const float*)d_in[b + 5]};
  };
  auto bn_at = [&](int b) {
    return Bn{(const float*)d_in[b], (const float*)d_in[b + 1],
              (const float*)d_in[b + 2], (const float*)d_in[b + 3]};
  };
  Conv c0 = conv_at(3), c1 = conv_at(13), c2 = conv_at(19);
  Bn   b0 = bn_at(9),   b1 = bn_at(25),  b2 = bn_at(29);
  const float* fcW = (const float*)d_in[33];
  const float* fcb = (const float*)d_in[34];

  float* ws = (float*)d_ws;
  size_t off = 0;
  auto alloc = [&](size_t n) {
    float* p = ws + off;
    off += (n + 63) & ~(size_t)63;
    return p;
  };
  float*    hbuf  = alloc((size_t)N_NODESK * HCK);
  float*    accb  = alloc((size_t)N_NODESK * HCK);
  float*    x0b   = alloc((size_t)N_NODESK * HCK);
  float*    xrb   = alloc((size_t)N_NODESK * HCK);
  float*    a_src = alloc((size_t)N_NODESK * HEADSK);
  float*    a_dst = alloc((size_t)N_NODESK * HEADSK);
  unsigned* amax  = (unsigned*)alloc((size_t)N_NODESK * HEADSK);
  float*    denom = alloc((size_t)N_NODESK * HEADSK);
  float*    alpha = alloc((size_t)ETOTK * HEADSK);
  float*    scal  = alloc(64);

  hipMemsetAsync(scal, 0, 64 * sizeof(float), stream);
  edge_sum_kernel<<<512, 256, 0, stream>>>(ew, scal + 8);

  const int gemmBlocks = ((N_NODESK / 16) + 7) / 8;  // 391
  const int edgeBlocks = (ETOTK + 255) / 256;
  const int edge4Blocks = (ETOTK * HEADSK + 255) / 256;
  const int nodeBlocks = (N_NODESK * HCK + 255) / 256;

  auto run_conv = [&](const float* xin, const Conv& cp, const Bn& bp,
                      int use_mean, float* xout) {
    gemm_wmma_kernel<HCK, false><<<gemmBlocks, 256, 0, stream>>>(
        xin, cp.W, nullptr, hbuf, N_NODESK);
    prep_kernel<<<1, 128, 0, stream>>>(cp.W_edge, cp.att_edge, scal);
    hipMemsetAsync(amax, 0, (size_t)N_NODESK * HEADSK * sizeof(unsigned), stream);
    hipMemsetAsync(denom, 0, (size_t)N_NODESK * HEADSK * sizeof(float), stream);
    hipMemsetAsync(accb, 0, (size_t)N_NODESK * HCK * sizeof(float), stream);
    node_att_kernel<<<(N_NODESK * HEADSK + 255) / 256, 256, 0, stream>>>(
        hbuf, cp.att_src, cp.att_dst, a_src, a_dst);
    edge_alpha_kernel<<<edgeBlocks, 256, 0, stream>>>(
        srcs, dsts, ew, scal, scal + 8, 1.f / N_EDGESK, use_mean, a_src, a_dst,
        alpha, amax);
    edge_exp_kernel<<<edgeBlocks, 256, 0, stream>>>(dsts, alpha, amax, denom);
    edge_scatter_kernel<<<edge4Blocks, 256, 0, stream>>>(srcs, dsts, alpha,
                                                         denom, hbuf, accb);
    finalize_kernel<<<nodeBlocks, 256, 0, stream>>>(
        accb, cp.bias, bp.gamma, bp.beta, bp.rm, bp.rv, xout);
  };

  run_conv(x,   c0, b0, /*use_mean=*/0, x0b);   // branch 0
  run_conv(x,   c1, b1, /*use_mean=*/1, xrb);   // layer 1 (input is x again)
  run_conv(xrb, c2, b2, /*use_mean=*/1, accb);  // layer 2 (finalize alias-safe)

  combine_kernel<<<nodeBlocks, 256, 0, stream>>>(accb, x0b, hbuf);  // relu(xr+x0)

  gemm_wmma_kernel<HIDK, true><<<gemmBlocks, 256, 0, stream>>>(
      hbuf, fcW, fcb, (float*)d_out, N_NODESK);
}